// PINNPhysicsLoss_4277787426962
// MI455X (gfx1250) — compile-verified
//
#include <hip/hip_runtime.h>
#include <hip/hip_bf16.h>

// Problem constants (fixed by the reference's setup_inputs()).
#define BB   16      // batch
#define NN   1024    // keypoints per batch
#define IMG_H 1024
#define IMG_W 1280
#define PTCH 11
#define KNN  5
#define BIGV 1.0e9f

typedef __attribute__((ext_vector_type(2))) float v2f;
typedef __attribute__((ext_vector_type(8))) float v8f;

// d_ws accumulator layout (floats):
//  [0] nconf  [1] epi_sum  [2] photo_sum
//  [3..18]  cnt[b]      [19..34] smooth_sum[b]
//  [35..50] slope_sum[b][51..66] zsum[b]
#define ACC_NCONF 0
#define ACC_EPI   1
#define ACC_PHOTO 2
#define ACC_CNT   3
#define ACC_SM    19
#define ACC_SL    35
#define ACC_ZS    51

__device__ __forceinline__ float huber(float d, float beta) {
    float ad = fabsf(d);
    return (ad < beta) ? (0.5f * ad * ad / beta) : (ad - 0.5f * beta);
}

__global__ void zero_kernel(float* acc) {
    acc[threadIdx.x] = 0.0f;   // 128 threads, zero the accumulator block
}

// ---------------- per-keypoint preprocessing -------------------------------
__global__ void preprocess_kernel(const float* __restrict__ kpl, const float* __restrict__ kpr,
                                  const float* __restrict__ scores, const float* __restrict__ Q,
                                  float* __restrict__ xA, float* __restrict__ zA,
                                  float* __restrict__ r2A, float* __restrict__ zhA,
                                  float* __restrict__ vfA, float* __restrict__ acc) {
    int tid = blockIdx.x * blockDim.x + threadIdx.x;
    if (tid >= BB * NN) return;
    int b = tid / NN;

    float xl = kpl[tid * 2 + 0], yl = kpl[tid * 2 + 1];
    float xr = kpr[tid * 2 + 0], yr = kpr[tid * 2 + 1];
    float s  = scores[tid];
    float mconf = (s > 0.1f) ? 1.0f : 0.0f;

    atomicAdd(&acc[ACC_NCONF], mconf);
    atomicAdd(&acc[ACC_EPI],   huber(fabsf(yl - yr), 1.0f) * mconf);

    // proj = p4 @ Q^T, p4 = (xl, yl, disp, 1)
    float disp = xl - xr;
    const float* q = Q + b * 16;
    float proj[4];
#pragma unroll
    for (int e = 0; e < 4; ++e)
        proj[e] = q[e * 4 + 0] * xl + q[e * 4 + 1] * yl + q[e * 4 + 2] * disp + q[e * 4 + 3];
    float Wc = fmaxf(proj[3], 1e-6f);
    float px = proj[0] / Wc, py = proj[1] / Wc, pz = proj[2] / Wc;

    float vf = (pz > 500.0f && pz < 30000.0f && s > 0.1f) ? 1.0f : 0.0f;
    float xm = px * 1e-3f, ym = py * 1e-3f, zm = pz * 1e-3f;

    xA[tid]  = xm;                 // WMMA A/B operand components:
    zA[tid]  = zm;                 //   a_i = (x, z, x^2+z^2, 1)
    r2A[tid] = xm * xm + zm * zm;  //   b_j = (-2x, -2z, 1, x^2+z^2)
    zhA[tid] = ym;                 // height (for smooth/slope)
    vfA[tid] = vf;

    atomicAdd(&acc[ACC_CNT + b], vf);
    atomicAdd(&acc[ACC_ZS  + b], ym * vf);
}

// ---------------- photometric loss ----------------------------------------
__device__ __forceinline__ float bilin(const float* __restrict__ img,
                                       float kx, float ky, int offx, int offy) {
    float x = fminf(fmaxf(kx + (float)offx, 0.0f), (float)(IMG_W - 1));
    float y = fminf(fmaxf(ky + (float)offy, 0.0f), (float)(IMG_H - 1));
    float x0 = floorf(x), y0 = floorf(y);
    int x0i = (int)x0, y0i = (int)y0;
    int x1i = min(x0i + 1, IMG_W - 1), y1i = min(y0i + 1, IMG_H - 1);
    float wx = x - x0, wy = y - y0;
    float v00 = img[y0i * IMG_W + x0i];
    float v01 = img[y0i * IMG_W + x1i];
    float v10 = img[y1i * IMG_W + x0i];
    float v11 = img[y1i * IMG_W + x1i];
    return v00 * (1.f - wx) * (1.f - wy) + v01 * wx * (1.f - wy)
         + v10 * (1.f - wx) * wy + v11 * wx * wy;
}

__global__ void photo_kernel(const float* __restrict__ left, const float* __restrict__ right,
                             const float* __restrict__ kpl, const float* __restrict__ kpr,
                             const float* __restrict__ scores, float* __restrict__ acc) {
    int kp = blockIdx.x;              // one block per keypoint
    float s = scores[kp];
    if (s <= 0.1f) return;            // uniform branch: zero contribution
    int b = kp / NN;
    float kxl = kpl[kp * 2], kyl = kpl[kp * 2 + 1];
    float kxr = kpr[kp * 2], kyr = kpr[kp * 2 + 1];
    const float* L = left  + (size_t)b * IMG_H * IMG_W;
    const float* R = right + (size_t)b * IMG_H * IMG_W;

    int t = threadIdx.x;
    if (t == 0) {   // warm the caches for the patch neighborhoods (global_prefetch_b8)
        __builtin_prefetch(L + (int)kyl * IMG_W + (int)kxl, 0, 1);
        __builtin_prefetch(R + (int)kyr * IMG_W + (int)kxr, 0, 1);
    }

    float diff = 0.0f;
    if (t < PTCH * PTCH) {
        int offy = t / PTCH - PTCH / 2;
        int offx = t % PTCH - PTCH / 2;
        diff = fabsf(bilin(L, kxl, kyl, offx, offy) - bilin(R, kxr, kyr, offx, offy));
    }
    __shared__ float red[128];
    red[t] = diff;
    __syncthreads();
#pragma unroll
    for (int off = 64; off > 0; off >>= 1) {
        if (t < off) red[t] += red[t + off];
        __syncthreads();
    }
    if (t == 0) atomicAdd(&acc[ACC_PHOTO], red[0] * (1.0f / (PTCH * PTCH)));
}

// ---------------- async global -> LDS staging (CDNA5 ASYNCcnt path) --------
// For an address in the LDS aperture, the LDS address is the low 32 bits (ISA 10.2), so the
// low 32 bits of a generic __shared__ pointer are the LDS byte offset the
// async-load instruction takes in its VDST operand.
__device__ __forceinline__ void async_copy_b128(float* lds_dst, const float* gsrc) {
    unsigned loff = (unsigned)(size_t)lds_dst;
    asm volatile("global_load_async_to_lds_b128 %0, %1, off"
                 :: "v"(loff), "v"(gsrc) : "memory");
}
__device__ __forceinline__ void async_wait0() {
    asm volatile("s_wait_asynccnt 0x0" ::: "memory");
}

// ---------------- kNN via WMMA distance tiles ------------------------------
// dist2[i][j] = (x_i,z_i,r_i,1) . (-2x_j,-2z_j,1,r_j) computed with
// V_WMMA_F32_16X16X4_F32: one 16x16 distance-matrix tile per instruction.
// Block = 128 threads (4 waves); each wave owns a 16-row strip. All per-batch
// operand arrays are staged into LDS once (async DMA), so the 64-iteration
// WMMA loop touches only LDS.
__global__ void knn_kernel(const float* __restrict__ xA, const float* __restrict__ zA,
                           const float* __restrict__ r2A, const float* __restrict__ zhA,
                           const float* __restrict__ vfA, float* __restrict__ acc) {
    __shared__ float sx[NN], sz[NN], sr[NN], sh[NN], sv[NN];   // 20 KB staged operands
    __shared__ float tiles[4 * 16 * 17];                       // per-wave padded tiles

    int blk  = blockIdx.x;
    int b    = blk >> 4;                 // / 16
    int t    = threadIdx.x;
    int w    = t >> 5;                   // wave id 0..3
    int lane = t & 31;
    int half = lane >> 4;
    int l16  = lane & 15;
    int m_base = (blk & 15) * 64 + w * 16;

    const float* xb = xA  + b * NN;
    const float* zb = zA  + b * NN;
    const float* rb = r2A + b * NN;
    const float* hb = zhA + b * NN;
    const float* vb = vfA + b * NN;

    // Cooperative async staging: 128 threads x 16B chunks, 2 sweeps per array.
    for (int i = t; i < NN / 4; i += 128) {
        async_copy_b128(&sx[i * 4], xb + i * 4);
        async_copy_b128(&sz[i * 4], zb + i * 4);
        async_copy_b128(&sr[i * 4], rb + i * 4);
        async_copy_b128(&sh[i * 4], hb + i * 4);
        async_copy_b128(&sv[i * 4], vb + i * 4);
    }
    async_wait0();
    __syncthreads();

    float* tile = tiles + w * 16 * 17;

    // A 16x4 f32 operand (loop invariant): lanes 0-15 hold (K0,K1)=(x,z) for
    // row M=lane, lanes 16-31 hold (K2,K3)=(r2,1) for row M=lane-16.
    // Unconditional LDS loads + cndmask select keeps EXEC full (no branches).
    int mrow = m_base + l16;
    float mx = sx[mrow], mz = sz[mrow], mr = sr[mrow];
    v2f a;
    a.x = half ? mr : mx;
    a.y = half ? 1.0f : mz;

    float dk[KNN];
    int   ik[KNN];
#pragma unroll
    for (int k = 0; k < KNN; ++k) { dk[k] = 3.0e38f; ik[k] = 0; }

    for (int j = 0; j < NN / 16; ++j) {
        int col = j * 16 + l16;
        float cx = sx[col], cz = sz[col], cr = sr[col], cv = sv[col];
        // B 4x16 operand: lanes 0-15 hold rows (K0,K1)=(-2x,-2z) for col
        // N=lane, lanes 16-31 hold rows (K2,K3)=(1,r2).
        v2f bm;
        bm.x = half ? 1.0f : (-2.0f * cx);
        bm.y = half ? cr   : (-2.0f * cz);

        v8f c = {0.f, 0.f, 0.f, 0.f, 0.f, 0.f, 0.f, 0.f};
        v8f d = __builtin_amdgcn_wmma_f32_16x16x4_f32(
            /*neg_a=*/false, a, /*neg_b=*/false, bm,
            /*c_mod=*/(short)0, c, /*reuse_a=*/false, /*reuse_b=*/false);

        bool colvalid = cv > 0.5f;
        // D layout: lane<16 -> rows m_base+r, lane>=16 -> rows m_base+8+r.
#pragma unroll
        for (int r = 0; r < 8; ++r) {
            int row_g = m_base + half * 8 + r;
            float v = d[r];
            if (!colvalid || row_g == col) v = BIGV;   // mask invalid cols + diagonal
            tile[(half * 8 + r) * 17 + l16] = v;
        }
        __syncthreads();   // all 4 waves iterate in lockstep (same trip count)
        if (lane < 16) {                    // lane owns row m_base+lane
#pragma unroll
            for (int c2 = 0; c2 < 16; ++c2) {
                float v  = tile[lane * 17 + c2];
                int   id = j * 16 + c2;
                if (v < dk[KNN - 1]) {      // sorted-ascending insert into top-5
                    dk[KNN - 1] = v; ik[KNN - 1] = id;
#pragma unroll
                    for (int k = KNN - 1; k > 0; --k) {
                        if (dk[k] < dk[k - 1]) {
                            float td = dk[k]; dk[k] = dk[k - 1]; dk[k - 1] = td;
                            int   ti = ik[k]; ik[k] = ik[k - 1]; ik[k - 1] = ti;
                        }
                    }
                }
            }
        }
        __syncthreads();
    }

    if (lane < 16) {
        int row = m_base + lane;
        float zr  = sh[row];
        float vfr = sv[row];
        if (vfr > 0.5f) {
            float meanh = 0.0f, slope = 0.0f;
#pragma unroll
            for (int k = 0; k < KNN; ++k) {
                float nd = fmaxf(sqrtf(fmaxf(dk[k], 1e-12f)), 0.001f);
                float nh = sh[ik[k]];       // neighbor-height gather from LDS
                meanh += nh;
                slope += fmaxf(fabsf(nh - zr) / nd - 0.4f, 0.0f);
            }
            meanh *= (1.0f / KNN);
            slope *= (1.0f / KNN);
            atomicAdd(&acc[ACC_SM + b], huber(zr - meanh, 0.01f));
            atomicAdd(&acc[ACC_SL + b], slope);
        }
    }
}

// ---------------- final scalar assembly ------------------------------------
__global__ void finalize_kernel(const float* __restrict__ acc, float* __restrict__ out) {
    if (threadIdx.x != 0) return;
    float nconf = fmaxf(acc[ACC_NCONF], 1.0f);
    float l_photo = acc[ACC_PHOTO] / nconf;
    float l_epi   = acc[ACC_EPI]   / nconf;
    float sm = 0.f, sl = 0.f, zm = 0.f, nb = 0.f;
    for (int b = 0; b < BB; ++b) {
        float cnt   = acc[ACC_CNT + b];
        float denom = fmaxf(cnt, 1.0f);
        float bval  = (cnt >= 10.0f) ? 1.0f : 0.0f;
        sm += (acc[ACC_SM + b] / denom) * bval;
        sl += (acc[ACC_SL + b] / denom) * bval;
        zm += fabsf(acc[ACC_ZS + b] / denom) * bval;
        nb += bval;
    }
    nb = fmaxf(nb, 1.0f);
    out[0] = l_photo;
    out[1] = l_epi;
    out[2] = sm / nb;
    out[3] = sl / nb;
    out[4] = zm / nb;
}

extern "C" void kernel_launch(void* const* d_in, const int* in_sizes, int n_in,
                              void* d_out, int out_size, void* d_ws, size_t ws_size,
                              hipStream_t stream) {
    const float* left  = (const float*)d_in[0];
    const float* right = (const float*)d_in[1];
    const float* kpl   = (const float*)d_in[2];
    const float* kpr   = (const float*)d_in[3];
    const float* sc    = (const float*)d_in[4];
    const float* Q     = (const float*)d_in[5];
    float* out = (float*)d_out;

    float* acc = (float*)d_ws;          // 128 floats of accumulators
    float* xA  = acc + 128;             // then 5 arrays of B*N floats
    float* zA  = xA  + BB * NN;
    float* r2A = zA  + BB * NN;
    float* zhA = r2A + BB * NN;
    float* vfA = zhA + BB * NN;

    zero_kernel<<<1, 128, 0, stream>>>(acc);
    preprocess_kernel<<<(BB * NN + 255) / 256, 256, 0, stream>>>(
        kpl, kpr, sc, Q, xA, zA, r2A, zhA, vfA, acc);
    photo_kernel<<<BB * NN, 128, 0, stream>>>(left, right, kpl, kpr, sc, acc);
    knn_kernel<<<BB * 16, 128, 0, stream>>>(xA, zA, r2A, zhA, vfA, acc);
    finalize_kernel<<<1, 32, 0, stream>>>(acc, out);
}